// LUTLayer_25847113188161
// MI455X (gfx1250) — compile-verified
//
#include <hip/hip_runtime.h>
#include <math.h>

#define BATCH      1024
#define INPUT_DIM  4096
#define N_LUTS     16384
#define TABLE      16
#define NB         256            // threads per block
#define BT         8              // batch rows staged in LDS (8*4096*4 = 128 KiB)
#define LPT        8              // luts per thread
#define LPB        (NB * LPT)     // 2048 luts per block

#if defined(__has_builtin)
# if __has_builtin(__builtin_amdgcn_global_load_async_to_lds_b128)
#  define HAVE_ASYNC_B128 1
# endif
# if __has_builtin(__builtin_amdgcn_s_wait_asynccnt)
#  define HAVE_WAIT_ASYNC 1
# endif
#endif

typedef int v4i __attribute__((ext_vector_type(4)));
typedef __attribute__((address_space(1))) v4i* gv4i_p;   // global ptr to 16B
typedef __attribute__((address_space(3))) v4i* lv4i_p;   // LDS ptr to 16B

// Async DMA of 16 contiguous bytes global -> LDS (tracked by ASYNCcnt).
__device__ __forceinline__ void stage16(const float* g, float* l) {
#if defined(HAVE_ASYNC_B128)
  __builtin_amdgcn_global_load_async_to_lds_b128((gv4i_p)g, (lv4i_p)l, 0, 0);
#else
  unsigned lds_off = (unsigned)(unsigned long long)(lv4i_p)l;
  asm volatile("global_load_async_to_lds_b128 %0, %1, off"
               :: "v"(lds_off), "v"(g)
               : "memory");
#endif
}

__device__ __forceinline__ void wait_async0() {
#if defined(HAVE_WAIT_ASYNC)
  __builtin_amdgcn_s_wait_asynccnt(0);
#else
  asm volatile("s_wait_asynccnt 0x0" ::: "memory");
#endif
}

// sigma_w = softmax([w, w_comp], axis=0)[0] = sigmoid(w - w_comp), elementwise.
__global__ __launch_bounds__(NB) void sigma_kernel(const float* __restrict__ w,
                                                   const float* __restrict__ wc,
                                                   float* __restrict__ sw, int n) {
  int i = blockIdx.x * NB + threadIdx.x;
  if (i < n) {
    sw[i] = 1.0f / (1.0f + expf(wc[i] - w[i]));
  }
}

__global__ __launch_bounds__(NB) void lut_kernel(const float* __restrict__ x,
                                                 const float* __restrict__ sw,
                                                 const int*   __restrict__ idx,
                                                 float*       __restrict__ out) {
  extern __shared__ float xs[];            // BT * INPUT_DIM floats (128 KiB)

  const int b0 = blockIdx.y * BT;
  const int n0 = blockIdx.x * LPB;

  // Stage BT contiguous rows of x (one contiguous 128 KiB block) into LDS.
  const float* src = x + (size_t)b0 * INPUT_DIM;
  const int chunks_per_thread = (BT * INPUT_DIM) / (NB * 4);  // 32 x b128 each
  for (int i = 0; i < chunks_per_thread; ++i) {
    const int c = (threadIdx.x + i * NB) * 4;                 // float offset
    stage16(src + c, xs + c);
  }
  wait_async0();
  __syncthreads();

  for (int l = 0; l < LPT; ++l) {
    const int n = n0 + threadIdx.x + l * NB;                  // coalesced in n
    const int i0 = idx[0 * N_LUTS + n];
    const int i1 = idx[1 * N_LUTS + n];
    const int i2 = idx[2 * N_LUTS + n];
    const int i3 = idx[3 * N_LUTS + n];
    const float4* swv = (const float4*)(sw + (size_t)n * TABLE);
    const float4 s0 = swv[0];   // t = 0..3   (a0=0, a1=0)
    const float4 s1 = swv[1];   // t = 4..7   (a0=0, a1=1)
    const float4 s2 = swv[2];   // t = 8..11  (a0=1, a1=0)
    const float4 s3 = swv[3];   // t = 12..15 (a0=1, a1=1)
#pragma unroll
    for (int b = 0; b < BT; ++b) {
      const float* row = xs + b * INPUT_DIM;
      const float x0 = row[i0], x1 = row[i1], x2 = row[i2], x3 = row[i3];
      // probs[t] = prod_i ((t>>(3-i))&1 ? x_i : 1-x_i); fold sigma_w groups.
      const float y2 = 1.0f - x2, y3 = 1.0f - x3;
      const float q00 = y2 * y3, q01 = y2 * x3, q10 = x2 * y3, q11 = x2 * x3;
      const float d0 = fmaf(q11, s0.w, fmaf(q10, s0.z, fmaf(q01, s0.y, q00 * s0.x)));
      const float d1 = fmaf(q11, s1.w, fmaf(q10, s1.z, fmaf(q01, s1.y, q00 * s1.x)));
      const float d2 = fmaf(q11, s2.w, fmaf(q10, s2.z, fmaf(q01, s2.y, q00 * s2.x)));
      const float d3 = fmaf(q11, s3.w, fmaf(q10, s3.z, fmaf(q01, s3.y, q00 * s3.x)));
      const float y0 = 1.0f - x0, y1 = 1.0f - x1;
      const float e0 = fmaf(x1, d1, y1 * d0);
      const float e1 = fmaf(x1, d3, y1 * d2);
      out[(size_t)(b0 + b) * N_LUTS + n] = fmaf(x0, e1, y0 * e0);
    }
  }
}

extern "C" void kernel_launch(void* const* d_in, const int* in_sizes, int n_in,
                              void* d_out, int out_size, void* d_ws, size_t ws_size,
                              hipStream_t stream) {
  const float* x   = (const float*)d_in[0];   // (1024, 4096) f32
  const float* w   = (const float*)d_in[1];   // (16384, 16) f32
  const float* wc  = (const float*)d_in[2];   // (16384, 16) f32
  const int*   idx = (const int*)d_in[3];     // (4, 16384) i32
  float* out = (float*)d_out;                 // (1024, 16384) f32
  float* sw  = (float*)d_ws;                  // 16384*16 f32 = 1 MiB scratch

  const int nsw = N_LUTS * TABLE;
  sigma_kernel<<<(nsw + NB - 1) / NB, NB, 0, stream>>>(w, wc, sw, nsw);

  dim3 grid(N_LUTS / LPB, BATCH / BT);        // (8, 128)
  lut_kernel<<<grid, NB, BT * INPUT_DIM * sizeof(float), stream>>>(x, sw, idx, out);
}